// GNMT_89215060673325
// MI455X (gfx1250) — compile-verified
//
#include <hip/hip_runtime.h>
#include <hip/hip_bf16.h>

// ---------------------------------------------------------------------------
// GNMT decoder step on MI455X (gfx1250), fp32 via V_WMMA_F32_16X16X4_F32.
// ~353MB of fp32 weights streamed once => ~15us memory floor @ 23.3 TB/s.
// This revision: (a) register double-buffering so each wave keeps a full
// chunk of loads in flight (no s_wait_loadcnt 0 before every WMMA),
// (b) 1-wave blocks so the sequential GRU chain spreads across all WGPs.
// ---------------------------------------------------------------------------

#define B_  64
#define E_  1024
#define H_  1024
#define F_  2048
#define V_  32000
#define NL_ 8

typedef float v2f __attribute__((ext_vector_type(2)));
typedef float v8f __attribute__((ext_vector_type(8)));

__device__ __forceinline__ v8f wmma4(v2f a, v2f b, v8f c) {
  // D = A(16x4 f32) x B(4x16 f32) + C(16x16 f32)
  return __builtin_amdgcn_wmma_f32_16x16x4_f32(
      false, a, false, b, (short)0, c, false, false);
}

__device__ __forceinline__ v2f ld2(const float* p) { return *(const v2f*)p; }

__device__ __forceinline__ float sigmoidf_(float x) {
  return 1.0f / (1.0f + __expf(-x));
}
__device__ __forceinline__ float tanhf_(float x) {
  return 1.0f - 2.0f / (__expf(2.0f * x) + 1.0f);
}

// ---------------------------------------------------------------------------
// Embedding gather + relu
// ---------------------------------------------------------------------------
__global__ void embed_relu_kernel(const int* __restrict__ x,
                                  const float* __restrict__ emb,
                                  float* __restrict__ xe) {
  int i = blockIdx.x * blockDim.x + threadIdx.x;   // over B*E
  int b = i >> 10;
  int e = i & (E_ - 1);
  float v = emb[(long)x[b] * E_ + e];
  xe[i] = v > 0.0f ? v : 0.0f;
}

// ---------------------------------------------------------------------------
// GEMM: out[m][n] = act( sum_k (X1+X2?)[m][k] * W[n][k] + bias[n] ), M = 64.
// One wave per block. blockIdx.x = (coltile << 2) | rowtile.
// Double-buffered chunks of GU k-steps keep 2*GU..3*GU loads in flight.
// ---------------------------------------------------------------------------
template <bool HASX2, bool RELU>
__global__ __launch_bounds__(32)
void gemm_wmma_kernel(const float* __restrict__ X1,
                      const float* __restrict__ X2,
                      const float* __restrict__ W,
                      const float* __restrict__ bias,
                      float* __restrict__ out,
                      int K, int N) {
  constexpr int GU = 8;                 // k-steps per chunk (32 k values)
  const int lane = threadIdx.x & 31;
  const int n16  = lane & 15;
  const int kh   = lane >> 4;           // K sub-offset {0,2}
  const int wave = blockIdx.x & 3;      // row tile
  const int colbase = (blockIdx.x >> 2) * 16;

  const float* a1 = X1 + (size_t)(wave * 16 + n16) * K + 2 * kh;
  const float* a2 = HASX2 ? (X2 + (size_t)(wave * 16 + n16) * K + 2 * kh) : nullptr;
  const float* bw = W + (size_t)(colbase + n16) * K + 2 * kh;

  v2f a1c[GU], a2c[GU], bc[GU];
#pragma unroll
  for (int u = 0; u < GU; ++u) {
    a1c[u] = ld2(a1 + 4 * u);
    if (HASX2) a2c[u] = ld2(a2 + 4 * u);
    bc[u] = ld2(bw + 4 * u);
  }

  v8f acc = {};
  for (int k = 4 * GU; k < K; k += 4 * GU) {
    v2f a1n[GU], a2n[GU], bn[GU];
#pragma unroll
    for (int u = 0; u < GU; ++u) {
      a1n[u] = ld2(a1 + k + 4 * u);
      if (HASX2) a2n[u] = ld2(a2 + k + 4 * u);
      bn[u] = ld2(bw + k + 4 * u);
    }
#pragma unroll
    for (int u = 0; u < GU; ++u) {
      v2f a = HASX2 ? (a1c[u] + a2c[u]) : a1c[u];
      acc = wmma4(a, bc[u], acc);
    }
#pragma unroll
    for (int u = 0; u < GU; ++u) {
      a1c[u] = a1n[u];
      if (HASX2) a2c[u] = a2n[u];
      bc[u] = bn[u];
    }
  }
#pragma unroll
  for (int u = 0; u < GU; ++u) {
    v2f a = HASX2 ? (a1c[u] + a2c[u]) : a1c[u];
    acc = wmma4(a, bc[u], acc);
  }

  const int ncol = colbase + n16;
  const float bn0 = bias[ncol];
#pragma unroll
  for (int v = 0; v < 8; ++v) {
    int m = wave * 16 + v + 8 * kh;
    float val = acc[v] + bn0;
    if (RELU) val = val > 0.0f ? val : 0.0f;
    out[(size_t)m * N + ncol] = val;
  }
}

// ---------------------------------------------------------------------------
// Fused GRU cell: xin = X1(+X2), hbar = hid+attn; 6 accumulators per wave,
// 6 WMMAs per k-step. One wave per block; blockIdx.x = (coltile<<2)|rowtile.
// ---------------------------------------------------------------------------
template <bool HASX2>
__global__ __launch_bounds__(32)
void gru_cell_kernel(const float* __restrict__ X1,
                     const float* __restrict__ X2,
                     const float* __restrict__ hid,
                     const float* __restrict__ attn,
                     const float* __restrict__ Wih,
                     const float* __restrict__ Whh,
                     const float* __restrict__ bih,
                     const float* __restrict__ bhh,
                     float* __restrict__ hout,
                     float* __restrict__ hout2) {
  constexpr int GU = 4;                 // k-steps per chunk (16 k values)
  const int lane = threadIdx.x & 31;
  const int n16  = lane & 15;
  const int kh   = lane >> 4;
  const int wave = blockIdx.x & 3;
  const int colbase = (blockIdx.x >> 2) * 16;

  const size_t arow = (size_t)(wave * 16 + n16) * H_ + 2 * kh;
  const float* ax1 = X1 + arow;
  const float* ax2 = HASX2 ? (X2 + arow) : nullptr;
  const float* ah1 = hid + arow;
  const float* ah2 = attn + arow;

  const size_t wr = ((size_t)(0 * H_ + colbase + n16)) * H_ + 2 * kh;
  const size_t wz = ((size_t)(1 * H_ + colbase + n16)) * H_ + 2 * kh;
  const size_t wn = ((size_t)(2 * H_ + colbase + n16)) * H_ + 2 * kh;
  const float* w_ir = Wih + wr; const float* w_iz = Wih + wz; const float* w_in = Wih + wn;
  const float* w_hr = Whh + wr; const float* w_hz = Whh + wz; const float* w_hn = Whh + wn;

  v8f acc_ir = {}, acc_iz = {}, acc_in = {};
  v8f acc_hr = {}, acc_hz = {}, acc_hn = {};

  // chunk buffers: activation streams + 6 weight streams
  v2f x1c[GU], x2c[GU], h1c[GU], h2c[GU];
  v2f irc[GU], izc[GU], inc[GU], hrc[GU], hzc[GU], hnc[GU];
#pragma unroll
  for (int u = 0; u < GU; ++u) {
    x1c[u] = ld2(ax1 + 4 * u);
    if (HASX2) x2c[u] = ld2(ax2 + 4 * u);
    h1c[u] = ld2(ah1 + 4 * u);
    h2c[u] = ld2(ah2 + 4 * u);
    irc[u] = ld2(w_ir + 4 * u); izc[u] = ld2(w_iz + 4 * u); inc[u] = ld2(w_in + 4 * u);
    hrc[u] = ld2(w_hr + 4 * u); hzc[u] = ld2(w_hz + 4 * u); hnc[u] = ld2(w_hn + 4 * u);
  }

  for (int k = 4 * GU; k < H_; k += 4 * GU) {
    v2f x1n[GU], x2n[GU], h1n[GU], h2n[GU];
    v2f irn[GU], izn[GU], inn[GU], hrn[GU], hzn[GU], hnn[GU];
#pragma unroll
    for (int u = 0; u < GU; ++u) {
      x1n[u] = ld2(ax1 + k + 4 * u);
      if (HASX2) x2n[u] = ld2(ax2 + k + 4 * u);
      h1n[u] = ld2(ah1 + k + 4 * u);
      h2n[u] = ld2(ah2 + k + 4 * u);
      irn[u] = ld2(w_ir + k + 4 * u); izn[u] = ld2(w_iz + k + 4 * u); inn[u] = ld2(w_in + k + 4 * u);
      hrn[u] = ld2(w_hr + k + 4 * u); hzn[u] = ld2(w_hz + k + 4 * u); hnn[u] = ld2(w_hn + k + 4 * u);
    }
#pragma unroll
    for (int u = 0; u < GU; ++u) {
      v2f ax = HASX2 ? (x1c[u] + x2c[u]) : x1c[u];
      v2f ah = h1c[u] + h2c[u];
      acc_ir = wmma4(ax, irc[u], acc_ir);
      acc_iz = wmma4(ax, izc[u], acc_iz);
      acc_in = wmma4(ax, inc[u], acc_in);
      acc_hr = wmma4(ah, hrc[u], acc_hr);
      acc_hz = wmma4(ah, hzc[u], acc_hz);
      acc_hn = wmma4(ah, hnc[u], acc_hn);
    }
#pragma unroll
    for (int u = 0; u < GU; ++u) {
      x1c[u] = x1n[u];
      if (HASX2) x2c[u] = x2n[u];
      h1c[u] = h1n[u]; h2c[u] = h2n[u];
      irc[u] = irn[u]; izc[u] = izn[u]; inc[u] = inn[u];
      hrc[u] = hrn[u]; hzc[u] = hzn[u]; hnc[u] = hnn[u];
    }
  }
#pragma unroll
  for (int u = 0; u < GU; ++u) {
    v2f ax = HASX2 ? (x1c[u] + x2c[u]) : x1c[u];
    v2f ah = h1c[u] + h2c[u];
    acc_ir = wmma4(ax, irc[u], acc_ir);
    acc_iz = wmma4(ax, izc[u], acc_iz);
    acc_in = wmma4(ax, inc[u], acc_in);
    acc_hr = wmma4(ah, hrc[u], acc_hr);
    acc_hz = wmma4(ah, hzc[u], acc_hz);
    acc_hn = wmma4(ah, hnc[u], acc_hn);
  }

  const int ncol = colbase + n16;
  const float bir = bih[0 * H_ + ncol], biz = bih[1 * H_ + ncol], bin = bih[2 * H_ + ncol];
  const float bhr = bhh[0 * H_ + ncol], bhz = bhh[1 * H_ + ncol], bhn = bhh[2 * H_ + ncol];

#pragma unroll
  for (int v = 0; v < 8; ++v) {
    int m = wave * 16 + v + 8 * kh;
    float r  = sigmoidf_((acc_ir[v] + bir) + (acc_hr[v] + bhr));
    float z  = sigmoidf_((acc_iz[v] + biz) + (acc_hz[v] + bhz));
    float nn = tanhf_((acc_in[v] + bin) + r * (acc_hn[v] + bhn));
    float hb = hid[(size_t)m * H_ + ncol] + attn[(size_t)m * H_ + ncol];
    float h  = (1.0f - z) * nn + z * hb;
    hout[(size_t)m * H_ + ncol] = h;
    if (hout2) hout2[(size_t)m * H_ + ncol] = h;
  }
}

// ---------------------------------------------------------------------------
// log_softmax pass 1: per-row max and log(sum(exp)). One block per row.
// ---------------------------------------------------------------------------
__global__ void softmax_reduce_kernel(const float* __restrict__ logits,
                                      float* __restrict__ red) {
  const int row = blockIdx.x;
  const int tid = threadIdx.x;
  __shared__ float sm[256];

  float mx = -3.402823466e+38f;
  for (int i = tid; i < V_; i += 256) {
    float v = logits[(size_t)row * V_ + i];
    mx = v > mx ? v : mx;
  }
  sm[tid] = mx;
  __syncthreads();
  for (int s = 128; s > 0; s >>= 1) {
    if (tid < s) sm[tid] = sm[tid] > sm[tid + s] ? sm[tid] : sm[tid + s];
    __syncthreads();
  }
  mx = sm[0];
  __syncthreads();

  float sum = 0.0f;
  for (int i = tid; i < V_; i += 256) {
    sum += __expf(logits[(size_t)row * V_ + i] - mx);
  }
  sm[tid] = sum;
  __syncthreads();
  for (int s = 128; s > 0; s >>= 1) {
    if (tid < s) sm[tid] += sm[tid + s];
    __syncthreads();
  }
  if (tid == 0) {
    red[2 * row + 0] = mx;
    red[2 * row + 1] = __logf(sm[0]);
  }
}

// log_softmax pass 2: logp = logits - max - log(sum)
__global__ void softmax_fix_kernel(float* __restrict__ logits,
                                   const float* __restrict__ red) {
  size_t i = (size_t)blockIdx.x * blockDim.x + threadIdx.x;   // over B*V
  int row = (int)(i / V_);
  logits[i] = logits[i] - red[2 * row] - red[2 * row + 1];
}

// ---------------------------------------------------------------------------
extern "C" void kernel_launch(void* const* d_in, const int* in_sizes, int n_in,
                              void* d_out, int out_size, void* d_ws, size_t ws_size,
                              hipStream_t stream) {
  (void)in_sizes; (void)n_in; (void)out_size; (void)ws_size;

  const float* feature = (const float*)d_in[0];
  const int*   x       = (const int*)  d_in[1];
  const float* attn_in = (const float*)d_in[2];
  const float* hiddens = (const float*)d_in[3];
  const float* emb     = (const float*)d_in[4];
  const float* map_W   = (const float*)d_in[5];
  const float* map_b   = (const float*)d_in[6];
  const float* ai_W    = (const float*)d_in[7];
  const float* ai_b    = (const float*)d_in[8];
  const float* ah_W    = (const float*)d_in[9];
  const float* ah_b    = (const float*)d_in[10];
  const float* ao_W    = (const float*)d_in[11];
  const float* ao_b    = (const float*)d_in[12];
  const float* gru_Wih = (const float*)d_in[13];
  const float* gru_Whh = (const float*)d_in[14];
  const float* gru_bih = (const float*)d_in[15];
  const float* gru_bhh = (const float*)d_in[16];
  const float* out_W   = (const float*)d_in[17];
  const float* out_b   = (const float*)d_in[18];

  // workspace layout (floats)
  float* ws    = (float*)d_ws;
  float* xe    = ws;
  float* feat  = xe    + (size_t)B_ * E_;
  float* attnA = feat  + (size_t)B_ * H_;
  float* attnB = attnA + (size_t)B_ * H_;
  float* attnC = attnB + (size_t)B_ * H_;
  float* red   = attnC + (size_t)B_ * H_;

  // output layout: [logp B*V][h1 B*H][new_hiddens NL*B*H]
  float* logp = (float*)d_out;
  float* h1   = logp + (size_t)B_ * V_;
  float* nh   = h1   + (size_t)B_ * H_;

  // 1) xe = relu(emb[x])
  embed_relu_kernel<<<(B_ * E_) / 256, 256, 0, stream>>>(x, emb, xe);

  // 2) feat = relu(feature @ map_W.T + map_b)     [K=2048, N=1024]
  gemm_wmma_kernel<false, true><<<(H_ / 16) * 4, 32, 0, stream>>>(
      feature, nullptr, map_W, map_b, feat, F_, H_);
  // 3) attn = relu((feat+attention) @ ai_W.T + ai_b)
  gemm_wmma_kernel<true, true><<<(H_ / 16) * 4, 32, 0, stream>>>(
      feat, attn_in, ai_W, ai_b, attnA, H_, H_);
  // 4) attn = relu(attn @ ah_W.T + ah_b)
  gemm_wmma_kernel<false, true><<<(H_ / 16) * 4, 32, 0, stream>>>(
      attnA, nullptr, ah_W, ah_b, attnB, H_, H_);
  // 5) attn = relu(attn @ ao_W.T + ao_b)
  gemm_wmma_kernel<false, true><<<(H_ / 16) * 4, 32, 0, stream>>>(
      attnB, nullptr, ao_W, ao_b, attnC, H_, H_);

  const size_t wih_l = (size_t)3 * H_ * E_;
  const size_t whh_l = (size_t)3 * H_ * H_;
  const size_t b_l   = (size_t)3 * H_;
  const size_t hh    = (size_t)B_ * H_;

  const float* cx1[NL_] = { xe,        nh,        nh + hh,   nh + hh,
                            nh + 2*hh, nh + 3*hh, nh + 4*hh, nh + 5*hh };
  const float* cx2[NL_] = { nullptr,   nullptr,   nullptr,   nh + 2*hh,
                            nh + 3*hh, nh + 4*hh, nh + 5*hh, nh + 6*hh };

  for (int i = 0; i < NL_; ++i) {
    const float* hid_i = hiddens + (size_t)i * hh;
    const float* Wih_i = gru_Wih + (size_t)i * wih_l;
    const float* Whh_i = gru_Whh + (size_t)i * whh_l;
    const float* bih_i = gru_bih + (size_t)i * b_l;
    const float* bhh_i = gru_bhh + (size_t)i * b_l;
    float* hout_i  = nh + (size_t)i * hh;
    float* hout2_i = (i == 0) ? h1 : nullptr;
    if (i >= 3) {
      gru_cell_kernel<true><<<(H_ / 16) * 4, 32, 0, stream>>>(
          cx1[i], cx2[i], hid_i, attnC, Wih_i, Whh_i, bih_i, bhh_i,
          hout_i, hout2_i);
    } else {
      gru_cell_kernel<false><<<(H_ / 16) * 4, 32, 0, stream>>>(
          cx1[i], cx2[i], hid_i, attnC, Wih_i, Whh_i, bih_i, bhh_i,
          hout_i, hout2_i);
    }
  }

  // logits = x8 @ out_W.T + out_b   [K=1024, N=32000] -> logp region (raw)
  gemm_wmma_kernel<false, false><<<(V_ / 16) * 4, 32, 0, stream>>>(
      nh + 7 * hh, nullptr, out_W, out_b, logp, H_, V_);

  // log_softmax in place
  softmax_reduce_kernel<<<B_, 256, 0, stream>>>(logp, red);
  softmax_fix_kernel<<<(B_ * V_) / 256, 256, 0, stream>>>(logp, red);
}